// TSCH_NN_75960791597964
// MI455X (gfx1250) — compile-verified
//
#include <hip/hip_runtime.h>
#include <math.h>

typedef __attribute__((ext_vector_type(2))) float v2f;
typedef __attribute__((ext_vector_type(8))) float v8f;

// ---------------------------------------------------------------------------
// utility
// ---------------------------------------------------------------------------
__global__ void fillf_kernel(float* __restrict__ p, float v, int n) {
    int i = blockIdx.x * blockDim.x + threadIdx.x;
    if (i < n) p[i] = v;
}

__device__ __forceinline__ void atomicMaxF(float* addr, float val) {
    // non-negative floats order like signed ints; negative floats
    // reverse-order like unsigned ints.
    if (val >= 0.0f)
        atomicMax((int*)addr, __float_as_int(val));
    else
        atomicMin((unsigned int*)addr, __float_as_uint(val));
}

// ---------------------------------------------------------------------------
// w_comb[j] = sum_f W_edge[j][f] * att_edge[f]        (De x 16) @ (16)
// ---------------------------------------------------------------------------
__global__ void wcomb_kernel(const float* __restrict__ W_edge,
                             const float* __restrict__ att_edge,
                             float* __restrict__ wcomb, int De) {
    int j = threadIdx.x;
    if (j < De) {
        float s = 0.0f;
        #pragma unroll
        for (int f = 0; f < 16; ++f) s += W_edge[j * 16 + f] * att_edge[f];
        wcomb[j] = s;
    }
}

// ---------------------------------------------------------------------------
// h = x @ W : [N,20]@[20,16] via V_WMMA_F32_16X16X4_F32, one wave32 per
// 16-node tile (5 K-chunks). Fused: a_src[i]=h[i,:].att_src, a_dst likewise,
// via shfl_xor reductions over the 16-lane halves holding each C-row.
// A frag (16x4): lane<16 -> M=lane, K={k0,k0+1}; lane>=16 -> K={k0+2,k0+3}
// B frag (4x16): vgpr j -> K = k0+2*half+j, N = lane&15
// C/D (16x16):   vgpr r -> M = r+8*half,    N = lane&15
// ---------------------------------------------------------------------------
__global__ void node_gemm_att_kernel(const float* __restrict__ x,   // [N,20]
                                     const float* __restrict__ W,   // [20,16]
                                     const float* __restrict__ att_src,
                                     const float* __restrict__ att_dst,
                                     float* __restrict__ h,         // [N,16]
                                     float* __restrict__ as,        // [N]
                                     float* __restrict__ ad,        // [N]
                                     int nTiles) {
    int wave = threadIdx.x >> 5;
    int lane = threadIdx.x & 31;
    int tile = blockIdx.x * (blockDim.x >> 5) + wave;
    if (tile >= nTiles) return;  // wave-uniform: EXEC stays all-ones

    const int half = lane >> 4;
    const int l16  = lane & 15;
    const int row  = tile * 16 + l16;

    v8f c = {};
    #pragma unroll
    for (int k0 = 0; k0 < 20; k0 += 4) {
        v2f a, b;
        int ka = k0 + 2 * half;
        a.x = x[(size_t)row * 20 + ka + 0];
        a.y = x[(size_t)row * 20 + ka + 1];
        b.x = W[(ka + 0) * 16 + l16];
        b.y = W[(ka + 1) * 16 + l16];
        c = __builtin_amdgcn_wmma_f32_16x16x4_f32(false, a, false, b,
                                                  (short)0, c, false, false);
    }

    // store h tile
    #pragma unroll
    for (int r = 0; r < 8; ++r) {
        int m = r + 8 * half;
        h[(size_t)(tile * 16 + m) * 16 + l16] = c[r];
    }

    // fused per-row attention dots: reduce across the 16 lanes of each half
    float asv = att_src[l16];
    float adv = att_dst[l16];
    #pragma unroll
    for (int r = 0; r < 8; ++r) {
        float vs = c[r] * asv;
        float vd = c[r] * adv;
        #pragma unroll
        for (int o = 1; o < 16; o <<= 1) {
            vs += __shfl_xor(vs, o, 32);
            vd += __shfl_xor(vd, o, 32);
        }
        if (l16 == 0) {
            int rw = tile * 16 + r + 8 * half;
            as[rw] = vs;
            ad[rw] = vd;
        }
    }
}

// ---------------------------------------------------------------------------
// fused = relu([out_i | out_t] @ W_fuse + b_fuse) : [N,32]@[32,16], WMMA
// ---------------------------------------------------------------------------
__global__ void fuse_gemm16_kernel(const float* __restrict__ oi,  // [N,16]
                                   const float* __restrict__ ot,  // [N,16]
                                   const float* __restrict__ Wf,  // [32,16]
                                   const float* __restrict__ bf,  // [16]
                                   float* __restrict__ fused,     // [N,16]
                                   int nTiles) {
    int wave = threadIdx.x >> 5;
    int lane = threadIdx.x & 31;
    int tile = blockIdx.x * (blockDim.x >> 5) + wave;
    if (tile >= nTiles) return;

    const int half = lane >> 4;
    const int l16  = lane & 15;
    const int row  = tile * 16 + l16;

    v8f c = {};
    #pragma unroll
    for (int k0 = 0; k0 < 32; k0 += 4) {
        int ka = k0 + 2 * half;  // ka, ka+1 always in the same source half
        v2f a, b;
        if (ka < 16) {
            a.x = oi[(size_t)row * 16 + ka + 0];
            a.y = oi[(size_t)row * 16 + ka + 1];
        } else {
            a.x = ot[(size_t)row * 16 + (ka - 16) + 0];
            a.y = ot[(size_t)row * 16 + (ka - 16) + 1];
        }
        b.x = Wf[(ka + 0) * 16 + l16];
        b.y = Wf[(ka + 1) * 16 + l16];
        c = __builtin_amdgcn_wmma_f32_16x16x4_f32(false, a, false, b,
                                                  (short)0, c, false, false);
    }
    float bias = bf[l16];
    #pragma unroll
    for (int r = 0; r < 8; ++r) {
        int m = r + 8 * half;
        float v = c[r] + bias;
        fused[(size_t)(tile * 16 + m) * 16 + l16] = v > 0.0f ? v : 0.0f;
    }
}

// ---------------------------------------------------------------------------
// edge pass 1: stage 256 edge_attr rows through LDS (coalesced HBM reads),
// score = leaky_relu(a_src[src]+a_dst[dst]+ea.wcomb, 0.2), atomic-max into
// m[dst]; block-reduced accumulation of sum(a_edge) for the self-loop mean.
// ---------------------------------------------------------------------------
__global__ void edge_score_kernel(const float* __restrict__ ea,  // [E,De]
                                  const int* __restrict__ src,
                                  const int* __restrict__ dst,
                                  const float* __restrict__ as,
                                  const float* __restrict__ ad,
                                  const float* __restrict__ wcomb,
                                  float* __restrict__ score,
                                  float* __restrict__ m,
                                  float* __restrict__ sum_ae,
                                  int E, int De) {
    __shared__ float sattr[256 * 19];  // max De = 19 -> ~19.5 KB
    const int tid  = threadIdx.x;
    const int base = blockIdx.x * 256;
    const int cnt  = min(256, E - base);

    // coalesced stage of this block's rows
    for (int j = tid; j < cnt * De; j += 256)
        sattr[j] = ea[(size_t)base * De + j];
    __syncthreads();

    float local = 0.0f;
    if (tid < cnt) {
        int e = base + tid;
        float aev = 0.0f;
        for (int j = 0; j < De; ++j)
            aev += sattr[tid * De + j] * wcomb[j];   // stride coprime w/ banks
        local = aev;
        int s = src[e], d = dst[e];
        float sc = as[s] + ad[d] + aev;
        sc = sc > 0.0f ? sc : 0.2f * sc;
        score[e] = sc;
        atomicMaxF(&m[d], sc);
    }

    // block-reduce a_edge sum, one atomic per block (reuse LDS)
    __syncthreads();
    sattr[tid] = local;
    __syncthreads();
    for (int off = 128; off > 0; off >>= 1) {
        if (tid < off) sattr[tid] += sattr[tid + off];
        __syncthreads();
    }
    if (tid == 0) atomicAdd(sum_ae, sattr[0]);
}

// ---------------------------------------------------------------------------
// self-loop pass: finalize m, seed den with the self-loop exp weight
// ---------------------------------------------------------------------------
__global__ void selfloop_kernel(const float* __restrict__ as,
                                const float* __restrict__ ad,
                                const float* __restrict__ sum_ae, float invE,
                                float* __restrict__ m,
                                float* __restrict__ den,
                                float* __restrict__ sself, int N) {
    int i = blockIdx.x * blockDim.x + threadIdx.x;
    if (i >= N) return;
    float aemean = sum_ae[0] * invE;
    float s = as[i] + ad[i] + aemean;
    s = s > 0.0f ? s : 0.2f * s;
    float mv = fmaxf(m[i], s);
    m[i] = mv;
    sself[i] = s;
    den[i] = __expf(s - mv);
}

// ---------------------------------------------------------------------------
// edge pass 2 (merged numerator+denominator; normalization is linear):
//   w = exp(score - m[dst]);  den[dst] += w;  num[dst,:] += h[src,:] * w
// All target tables are L2-resident (6.4 MB << 192 MB).
// ---------------------------------------------------------------------------
__global__ void edge_accum_kernel(const int* __restrict__ src,
                                  const int* __restrict__ dst,
                                  const float* __restrict__ score,
                                  const float* __restrict__ m,
                                  float* __restrict__ den,
                                  const float* __restrict__ h,
                                  float* __restrict__ num, int E) {
    int e = blockIdx.x * blockDim.x + threadIdx.x;
    if (e >= E) return;
    int s = src[e], d = dst[e];
    float w = __expf(score[e] - m[d]);
    atomicAdd(&den[d], w);
    #pragma unroll
    for (int f = 0; f < 16; ++f)
        atomicAdd(&num[(size_t)d * 16 + f], h[(size_t)s * 16 + f] * w);
}

// ---------------------------------------------------------------------------
// node finalize: out = (num + h*w_self)/den + bias   (in place on num)
// ---------------------------------------------------------------------------
__global__ void node_fin_kernel(const float* __restrict__ h,
                                const float* __restrict__ sself,
                                const float* __restrict__ m,
                                const float* __restrict__ den,
                                const float* __restrict__ bias,
                                float* __restrict__ num, int N) {
    int i = blockIdx.x * blockDim.x + threadIdx.x;
    if (i >= N) return;
    float wself = __expf(sself[i] - m[i]);
    float rden  = 1.0f / den[i];
    #pragma unroll
    for (int f = 0; f < 16; ++f) {
        size_t idx = (size_t)i * 16 + f;
        num[idx] = (num[idx] + h[idx] * wself) * rden + bias[f];
    }
}

// ---------------------------------------------------------------------------
// global mean pool (accumulate) and heads
// ---------------------------------------------------------------------------
__global__ void pool_kernel(const float* __restrict__ fused,
                            const int* __restrict__ batch,
                            float* __restrict__ g, float* __restrict__ cnt,
                            int N) {
    int i = blockIdx.x * blockDim.x + threadIdx.x;
    if (i >= N) return;
    int b = batch[i];
    #pragma unroll
    for (int f = 0; f < 16; ++f)
        atomicAdd(&g[(size_t)b * 16 + f], fused[(size_t)i * 16 + f]);
    atomicAdd(&cnt[b], 1.0f);
}

__global__ void heads_kernel(const float* __restrict__ g,
                             const float* __restrict__ cnt,
                             const float* __restrict__ Wev,
                             const float* __restrict__ bev,
                             const float* __restrict__ Wenv,
                             const float* __restrict__ benv,
                             float* __restrict__ out, int G) {
    int gi = blockIdx.x * blockDim.x + threadIdx.x;
    if (gi >= G) return;
    float c = fmaxf(cnt[gi], 1.0f);
    float gm[16];
    #pragma unroll
    for (int f = 0; f < 16; ++f) gm[f] = g[(size_t)gi * 16 + f] / c;
    #pragma unroll
    for (int k = 0; k < 3; ++k) {
        float se = bev[k], sv = benv[k];
        #pragma unroll
        for (int f = 0; f < 16; ++f) {
            se += gm[f] * Wev[f * 3 + k];
            sv += gm[f] * Wenv[f * 3 + k];
        }
        out[gi * 3 + k] = se;               // out_event
        out[G * 3 + gi * 3 + k] = sv;       // out_env (concatenated)
    }
}

// ---------------------------------------------------------------------------
// launcher
// ---------------------------------------------------------------------------
extern "C" void kernel_launch(void* const* d_in, const int* in_sizes, int n_in,
                              void* d_out, int out_size, void* d_ws, size_t ws_size,
                              hipStream_t stream) {
    (void)n_in; (void)ws_size;

    const float* x      = (const float*)d_in[0];
    const int*   ei_i   = (const int*)d_in[1];
    const float* ea_i   = (const float*)d_in[2];
    const int*   ei_t   = (const int*)d_in[3];
    const float* ea_t   = (const float*)d_in[4];
    const int*   batch  = (const int*)d_in[5];
    const float* W_i    = (const float*)d_in[6];
    const float* asrc_i = (const float*)d_in[7];
    const float* adst_i = (const float*)d_in[8];
    const float* We_i   = (const float*)d_in[9];
    const float* ate_i  = (const float*)d_in[10];
    const float* bias_i = (const float*)d_in[11];
    const float* W_t    = (const float*)d_in[12];
    const float* asrc_t = (const float*)d_in[13];
    const float* adst_t = (const float*)d_in[14];
    const float* We_t   = (const float*)d_in[15];
    const float* ate_t  = (const float*)d_in[16];
    const float* bias_t = (const float*)d_in[17];
    const float* W_fuse = (const float*)d_in[18];
    const float* b_fuse = (const float*)d_in[19];
    const float* W_ev   = (const float*)d_in[20];
    const float* b_ev   = (const float*)d_in[21];
    const float* W_env  = (const float*)d_in[22];
    const float* b_env  = (const float*)d_in[23];

    const int N  = in_sizes[0] / 20;        // 100000
    const int Ei = in_sizes[2] / 19;        // 3200000
    const int Et = in_sizes[4] / 7;         // 3200000
    const int G  = out_size / 6;            // 256

    // --- workspace carve-up (floats) ---
    float* ws = (float*)d_ws;
    size_t off = 0;
    auto alloc = [&](size_t n) { float* p = ws + off; off += n; return p; };
    float* h_i   = alloc((size_t)N * 16);
    float* h_t   = alloc((size_t)N * 16);
    float* out_i = alloc((size_t)N * 16);   // num, then final conv output
    float* out_t = alloc((size_t)N * 16);
    float* fused = alloc((size_t)N * 16);
    float* as_i  = alloc(N);  float* ad_i = alloc(N);
    float* as_t  = alloc(N);  float* ad_t = alloc(N);
    float* m_i   = alloc(N);  float* m_t  = alloc(N);
    float* den_i = alloc(N);  float* den_t = alloc(N);
    float* ss_i  = alloc(N);  float* ss_t  = alloc(N);
    float* sc_i  = alloc(Ei); float* sc_t  = alloc(Et);
    float* wc_i  = alloc(32); float* wc_t  = alloc(32);
    float* sum_i = alloc(1);  float* sum_t = alloc(1);
    float* g     = alloc((size_t)G * 16);
    float* cnt   = alloc(G);

    const int TB = 256;
    auto nb = [](int n, int tb) { return (n + tb - 1) / tb; };
    const int nTiles = N / 16;                 // N = 100000 -> 6250 tiles
    const int wavesPerBlock = TB / 32;
    const int gemmBlocks = (nTiles + wavesPerBlock - 1) / wavesPerBlock;

    // --- init accumulators (ws is poisoned; re-init every call) ---
    fillf_kernel<<<nb(N * 16, TB), TB, 0, stream>>>(out_i, 0.0f, N * 16);
    fillf_kernel<<<nb(N * 16, TB), TB, 0, stream>>>(out_t, 0.0f, N * 16);
    fillf_kernel<<<nb(N, TB), TB, 0, stream>>>(m_i, -INFINITY, N);
    fillf_kernel<<<nb(N, TB), TB, 0, stream>>>(m_t, -INFINITY, N);
    fillf_kernel<<<1, 64, 0, stream>>>(sum_i, 0.0f, 1);
    fillf_kernel<<<1, 64, 0, stream>>>(sum_t, 0.0f, 1);
    fillf_kernel<<<nb(G * 16 + G, TB), TB, 0, stream>>>(g, 0.0f, G * 16 + G);

    // --- folded edge weight vectors ---
    wcomb_kernel<<<1, 32, 0, stream>>>(We_i, ate_i, wc_i, 19);
    wcomb_kernel<<<1, 32, 0, stream>>>(We_t, ate_t, wc_t, 7);

    // --- node transforms (WMMA) with fused attention dots ---
    node_gemm_att_kernel<<<gemmBlocks, TB, 0, stream>>>(x, W_i, asrc_i, adst_i,
                                                        h_i, as_i, ad_i, nTiles);
    node_gemm_att_kernel<<<gemmBlocks, TB, 0, stream>>>(x, W_t, asrc_t, adst_t,
                                                        h_t, as_t, ad_t, nTiles);

    const int* src_i = ei_i;            const int* dst_i = ei_i + Ei;
    const int* src_t = ei_t;            const int* dst_t = ei_t + Et;

    // --- edge pass 1: scores + per-dst max + a_edge mean accumulation ---
    edge_score_kernel<<<nb(Ei, TB), TB, 0, stream>>>(ea_i, src_i, dst_i, as_i, ad_i,
                                                     wc_i, sc_i, m_i, sum_i, Ei, 19);
    edge_score_kernel<<<nb(Et, TB), TB, 0, stream>>>(ea_t, src_t, dst_t, as_t, ad_t,
                                                     wc_t, sc_t, m_t, sum_t, Et, 7);

    // --- self loops: finalize max, seed denominators ---
    selfloop_kernel<<<nb(N, TB), TB, 0, stream>>>(as_i, ad_i, sum_i, 1.0f / (float)Ei,
                                                  m_i, den_i, ss_i, N);
    selfloop_kernel<<<nb(N, TB), TB, 0, stream>>>(as_t, ad_t, sum_t, 1.0f / (float)Et,
                                                  m_t, den_t, ss_t, N);

    // --- edge pass 2: merged numerator + denominator accumulation ---
    edge_accum_kernel<<<nb(Ei, TB), TB, 0, stream>>>(src_i, dst_i, sc_i, m_i,
                                                     den_i, h_i, out_i, Ei);
    edge_accum_kernel<<<nb(Et, TB), TB, 0, stream>>>(src_t, dst_t, sc_t, m_t,
                                                     den_t, h_t, out_t, Et);

    // --- normalize + self-loop contribution + bias ---
    node_fin_kernel<<<nb(N, TB), TB, 0, stream>>>(h_i, ss_i, m_i, den_i, bias_i,
                                                  out_i, N);
    node_fin_kernel<<<nb(N, TB), TB, 0, stream>>>(h_t, ss_t, m_t, den_t, bias_t,
                                                  out_t, N);

    // --- fusion MLP (WMMA) ---
    fuse_gemm16_kernel<<<gemmBlocks, TB, 0, stream>>>(out_i, out_t, W_fuse, b_fuse,
                                                      fused, nTiles);

    // --- mean pool + heads ---
    pool_kernel<<<nb(N, TB), TB, 0, stream>>>(fused, batch, g, cnt, N);
    heads_kernel<<<nb(G, TB), TB, 0, stream>>>(g, cnt, W_ev, b_ev, W_env, b_env,
                                               (float*)d_out, G);
}